// GCNModel_5952824672615
// MI455X (gfx1250) — compile-verified
//
#include <hip/hip_runtime.h>

#define N_NODES  50000
#define N_EDGES  800000
#define N_GRAPHS 2048
#define CH       128
#define BN_EPS   1e-5f

typedef float v2f __attribute__((ext_vector_type(2)));
typedef float v8f __attribute__((ext_vector_type(8)));

// ---------------- utility ----------------
__global__ void fill_kernel(float* __restrict__ p, float v, int n) {
    int i = blockIdx.x * blockDim.x + threadIdx.x;
    int stride = gridDim.x * blockDim.x;
    for (; i < n; i += stride) p[i] = v;
}

// deg[d] starts at 1.0 (self loop); add 1 per in-edge
__global__ void degree_kernel(const int* __restrict__ dst, float* __restrict__ deg, int e) {
    int i = blockIdx.x * blockDim.x + threadIdx.x;
    if (i < e) unsafeAtomicAdd(&deg[dst[i]], 1.0f);
}

__global__ void rsqrt_kernel(float* __restrict__ deg, int n) {
    int i = blockIdx.x * blockDim.x + threadIdx.x;
    if (i < n) deg[i] = rsqrtf(deg[i]);   // deg >= 1 always (self loop)
}

// ---------------- fp32 WMMA GEMM: C[M,128] = A[M,128] * B[128,128] ----------------
// One wave -> one 16x16 tile of C, K looped in steps of 4 via v_wmma_f32_16x16x4_f32.
// A lane layout (ISA 7.12.2, 32-bit A 16x4): lanes 0-15 hold M=lane, VGPR0=K0/VGPR1=K1;
// lanes 16-31 hold M=lane-16, VGPR0=K2/VGPR1=K3. B mirrors with N across lanes.
// C/D: lane<16 -> col=lane, rows m0+r; lane>=16 -> col=lane-16, rows m0+8+r.
__global__ void gemm_wmma_f32(const float* __restrict__ A, const float* __restrict__ B,
                              float* __restrict__ C, int M) {
    int wave = (blockIdx.x * blockDim.x + threadIdx.x) >> 5;
    int lane = threadIdx.x & 31;
    int mt = wave >> 3;           // 8 tiles along N (128/16)
    int nt = wave & 7;
    int m0 = mt * 16, n0 = nt * 16;
    if (m0 >= M) return;          // uniform per wave; grid sized exactly anyway
    int half = lane >> 4;         // 0: K0/K1, 1: K2/K3
    int l    = lane & 15;

    const v2f* arow = (const v2f*)(A + (size_t)(m0 + l) * CH);  // contiguous K pairs
    v8f c = {};
    for (int k = 0; k < CH; k += 4) {
        v2f a = arow[(k >> 1) + half];              // A[m0+l][k+2*half], A[..][k+2*half+1]
        v2f b;
        int kb = k + half * 2;
        b.x = B[(size_t)(kb    ) * CH + n0 + l];
        b.y = B[(size_t)(kb + 1) * CH + n0 + l];
        c = __builtin_amdgcn_wmma_f32_16x16x4_f32(false, a, false, b,
                                                  (short)0, c, false, false);
    }
    #pragma unroll
    for (int r = 0; r < 8; ++r)
        C[(size_t)(m0 + r + half * 8) * CH + (n0 + l)] = c[r];
}

// ---------------- edge aggregation: out[d] += xw[s] * dinv[s]*dinv[d] ----------------
// one wave per (edge or self-loop); each lane handles 4 channels as float4
__global__ void edge_agg_kernel(const int* __restrict__ src, const int* __restrict__ dst,
                                const float* __restrict__ xw, const float* __restrict__ dinv,
                                float* __restrict__ out, int e, int n) {
    int idx  = blockIdx.x * (blockDim.x >> 5) + (threadIdx.x >> 5);
    int lane = threadIdx.x & 31;
    if (idx >= e + n) return;
    int s, d;
    if (idx < e) { s = src[idx]; d = dst[idx]; }
    else         { s = d = idx - e; }              // self loop
    float w = dinv[s] * dinv[d];
    float4 v = ((const float4*)(xw + (size_t)s * CH))[lane];
    float* o = out + (size_t)d * CH + lane * 4;
    unsafeAtomicAdd(o + 0, v.x * w);
    unsafeAtomicAdd(o + 1, v.y * w);
    unsafeAtomicAdd(o + 2, v.z * w);
    unsafeAtomicAdd(o + 3, v.w * w);
}

// ---------------- BatchNorm (training-mode, biased var) ----------------
__global__ void bn_stats_kernel(const float* __restrict__ x, float* __restrict__ ssum,
                                float* __restrict__ ssq, int n) {
    int c = threadIdx.x;  // 128 threads = channels; block strides rows (coalesced 512B rows)
    float s = 0.f, q = 0.f;
    for (int r = blockIdx.x; r < n; r += gridDim.x) {
        float v = x[(size_t)r * CH + c];
        s += v; q += v * v;
    }
    unsafeAtomicAdd(&ssum[c], s);
    unsafeAtomicAdd(&ssq[c],  q);
}

__global__ void bn_finalize_kernel(const float* __restrict__ ssum, const float* __restrict__ ssq,
                                   const float* __restrict__ gamma, const float* __restrict__ beta,
                                   float* __restrict__ scale, float* __restrict__ shift, int n) {
    int c = threadIdx.x;
    float inv_n = 1.0f / (float)n;
    float mean  = ssum[c] * inv_n;
    float var   = ssq[c] * inv_n - mean * mean;      // biased variance
    float sc    = gamma[c] * rsqrtf(var + BN_EPS);
    scale[c] = sc;
    shift[c] = beta[c] - mean * sc;
}

__global__ void bn_apply_relu_kernel(float* __restrict__ x, const float* __restrict__ scale,
                                     const float* __restrict__ shift, int total) {
    int i = blockIdx.x * blockDim.x + threadIdx.x;
    int stride = gridDim.x * blockDim.x;
    for (; i < total; i += stride) {
        int c = i & (CH - 1);
        x[i] = fmaxf(x[i] * scale[c] + shift[c], 0.0f);
    }
}

// ---------------- global mean pool + output projection ----------------
__global__ void pool_kernel(const float* __restrict__ h, const int* __restrict__ batch,
                            float* __restrict__ psum, float* __restrict__ pcnt, int n) {
    int node = blockIdx.x * (blockDim.x >> 5) + (threadIdx.x >> 5);
    int lane = threadIdx.x & 31;
    if (node >= n) return;
    int g = batch[node];
    float4 v = ((const float4*)(h + (size_t)node * CH))[lane];
    float* o = psum + (size_t)g * CH + lane * 4;
    unsafeAtomicAdd(o + 0, v.x);
    unsafeAtomicAdd(o + 1, v.y);
    unsafeAtomicAdd(o + 2, v.z);
    unsafeAtomicAdd(o + 3, v.w);
    if (lane == 0) unsafeAtomicAdd(&pcnt[g], 1.0f);
}

__global__ void out_kernel(const float* __restrict__ psum, const float* __restrict__ pcnt,
                           const float* __restrict__ Wout, const float* __restrict__ bout,
                           float* __restrict__ out, int g_total) {
    int g    = blockIdx.x * (blockDim.x >> 5) + (threadIdx.x >> 5);
    int lane = threadIdx.x & 31;
    if (g >= g_total) return;
    float4 s = ((const float4*)(psum + (size_t)g * CH))[lane];
    float4 w = ((const float4*)Wout)[lane];
    float part = s.x * w.x + s.y * w.y + s.z * w.z + s.w * w.w;
    #pragma unroll
    for (int off = 16; off > 0; off >>= 1) part += __shfl_down(part, off);
    if (lane == 0) out[g] = part / fmaxf(pcnt[g], 1.0f) + bout[0];
}

// ---------------- host orchestration ----------------
extern "C" void kernel_launch(void* const* d_in, const int* in_sizes, int n_in,
                              void* d_out, int out_size, void* d_ws, size_t ws_size,
                              hipStream_t stream) {
    const float* x      = (const float*)d_in[0];
    const int*   eidx   = (const int*)d_in[1];          // [2,E] flat: src then dst
    const int*   batch  = (const int*)d_in[2];
    const float* W1     = (const float*)d_in[3];
    // b1 (d_in[4]) cancels inside BatchNorm -> skipped (mathematically exact)
    const float* gamma1 = (const float*)d_in[5];
    const float* beta1  = (const float*)d_in[6];
    const float* W2     = (const float*)d_in[7];
    // b2 (d_in[8]) cancels as well
    const float* gamma2 = (const float*)d_in[9];
    const float* beta2  = (const float*)d_in[10];
    const float* Wout   = (const float*)d_in[11];
    const float* bout   = (const float*)d_in[12];
    float* out = (float*)d_out;

    const int* src = eidx;
    const int* dst = eidx + N_EDGES;

    // workspace carve (floats)
    float* ws    = (float*)d_ws;
    float* buf0  = ws;                                   // N*CH  (xw)
    float* buf1  = buf0 + (size_t)N_NODES * CH;          // N*CH  (agg / h)
    float* dinv  = buf1 + (size_t)N_NODES * CH;          // N     (deg -> dinv in place)
    float* ssum  = dinv + N_NODES;                       // CH
    float* ssq   = ssum + CH;                            // CH
    float* scale = ssq + CH;                             // CH
    float* shift = scale + CH;                           // CH
    float* psum  = shift + CH;                           // G*CH
    float* pcnt  = psum + (size_t)N_GRAPHS * CH;         // G

    const int NH = N_NODES * CH;
    const int gemm_blocks = (N_NODES / 16) * 8 / 8;      // waves/8 per 256-thr block = 3125
    const int agg_blocks  = (N_EDGES + N_NODES + 7) / 8; // 8 waves per block

    // degrees -> dinv
    fill_kernel<<<256, 256, 0, stream>>>(dinv, 1.0f, N_NODES);
    degree_kernel<<<(N_EDGES + 255) / 256, 256, 0, stream>>>(dst, dinv, N_EDGES);
    rsqrt_kernel<<<(N_NODES + 255) / 256, 256, 0, stream>>>(dinv, N_NODES);

    // ----- layer 1 -----
    gemm_wmma_f32<<<gemm_blocks, 256, 0, stream>>>(x, W1, buf0, N_NODES);
    fill_kernel<<<1024, 256, 0, stream>>>(buf1, 0.0f, NH);
    edge_agg_kernel<<<agg_blocks, 256, 0, stream>>>(src, dst, buf0, dinv, buf1, N_EDGES, N_NODES);
    fill_kernel<<<1, 256, 0, stream>>>(ssum, 0.0f, 2 * CH);  // ssum+ssq contiguous
    bn_stats_kernel<<<512, CH, 0, stream>>>(buf1, ssum, ssq, N_NODES);
    bn_finalize_kernel<<<1, CH, 0, stream>>>(ssum, ssq, gamma1, beta1, scale, shift, N_NODES);
    bn_apply_relu_kernel<<<1024, 256, 0, stream>>>(buf1, scale, shift, NH);

    // ----- layer 2 -----
    gemm_wmma_f32<<<gemm_blocks, 256, 0, stream>>>(buf1, W2, buf0, N_NODES);
    fill_kernel<<<1024, 256, 0, stream>>>(buf1, 0.0f, NH);
    edge_agg_kernel<<<agg_blocks, 256, 0, stream>>>(src, dst, buf0, dinv, buf1, N_EDGES, N_NODES);
    fill_kernel<<<1, 256, 0, stream>>>(ssum, 0.0f, 2 * CH);
    bn_stats_kernel<<<512, CH, 0, stream>>>(buf1, ssum, ssq, N_NODES);
    bn_finalize_kernel<<<1, CH, 0, stream>>>(ssum, ssq, gamma2, beta2, scale, shift, N_NODES);
    bn_apply_relu_kernel<<<1024, 256, 0, stream>>>(buf1, scale, shift, NH);

    // ----- pool + projection -----
    fill_kernel<<<256, 256, 0, stream>>>(psum, 0.0f, N_GRAPHS * CH + N_GRAPHS); // psum+pcnt contiguous
    pool_kernel<<<(N_NODES + 7) / 8, 256, 0, stream>>>(buf1, batch, psum, pcnt, N_NODES);
    out_kernel<<<(N_GRAPHS + 7) / 8, 256, 0, stream>>>(psum, pcnt, Wout, bout, out, N_GRAPHS);
}